// Classifier_GAT_GAP_43765716746307
// MI455X (gfx1250) — compile-verified
//
#include <hip/hip_runtime.h>
#include <math.h>

// ---------------------------------------------------------------------------
// Classifier_GAT_GAP for MI455X (gfx1250, wave32, WMMA)
// ---------------------------------------------------------------------------

typedef float v2f __attribute__((ext_vector_type(2)));
typedef float v8f __attribute__((ext_vector_type(8)));

#define SLOPE 0.2f
#define NGRAPH 8
#define HID 64
#define NHEAD 2
#define DOUT 128 // NHEAD * HID

// --- native fp32 atomics ----------------------------------------------------
__device__ __forceinline__ void atomic_add_f32(float* p, float v) {
  __hip_atomic_fetch_add(p, v, __ATOMIC_RELAXED, __HIP_MEMORY_SCOPE_AGENT);
}

// float atomic max via signed/unsigned integer atomics (valid for all IEEE
// orderings incl. -inf init): positive values -> signed atomicMax on int bits,
// negative values -> unsigned atomicMin on uint bits.
__device__ __forceinline__ void atomic_max_f32(float* addr, float value) {
  if (value >= 0.0f) {
    atomicMax((int*)addr, __float_as_int(value));
  } else {
    atomicMin((unsigned int*)addr, __float_as_uint(value));
  }
}

// --- fill -------------------------------------------------------------------
__global__ void fill_kernel(float* __restrict__ p, float v, int n) {
  int i = blockIdx.x * blockDim.x + threadIdx.x;
  if (i < n) p[i] = v;
}

// --- GEMM: Y[N,128] = X[N,K] @ W[K,128]  via V_WMMA_F32_16X16X4_F32 ---------
// block = (32, 8): 8 waves, wave y owns column tile y (16 cols).
// blockIdx.x owns row tile (16 rows). K in {128, 64}, multiple of 4.
__global__ void gemm_fc_wmma(const float* __restrict__ X,
                             const float* __restrict__ W,
                             float* __restrict__ Y, int Nrows, int K) {
  const int lane = threadIdx.x;           // 0..31
  const int col0 = threadIdx.y * 16;      // column tile base
  const int row0 = blockIdx.x * 16;       // row tile base
  const int m    = lane & 15;
  const int koff = (lane < 16) ? 0 : 2;   // A/B K sub-offset per ISA layout

  int row = row0 + m;
  if (row >= Nrows) row = Nrows - 1;      // clamp (tail-safe; N%16==0 here)
  const float* xrow = X + (size_t)row * K;

  v8f acc = {};
  for (int k = 0; k < K; k += 4) {
    // A 16x4 f32: VGPR0 = K=koff, VGPR1 = K=koff+1 (consecutive -> float2 load)
    v2f a;
    const float2 av = *(const float2*)(xrow + k + koff);
    a.x = av.x;
    a.y = av.y;
    // B 4x16 f32: VGPR j, lane l -> K = j + koff, Ncol = l%16
    v2f b;
    b.x = W[(size_t)(k + koff) * DOUT + col0 + m];
    b.y = W[(size_t)(k + koff + 1) * DOUT + col0 + m];
    acc = __builtin_amdgcn_wmma_f32_16x16x4_f32(
        /*neg_a=*/false, a, /*neg_b=*/false, b,
        /*c_mod=*/(short)0, acc, /*reuse_a=*/false, /*reuse_b=*/false);
  }

  // C/D 16x16 f32: VGPR v, lane l -> M = v + (l<16?0:8), Ncol = l%16
  const int mbase = (lane < 16) ? 0 : 8;
  for (int v = 0; v < 8; ++v) {
    const int r = row0 + mbase + v;
    if (r < Nrows) Y[(size_t)r * DOUT + col0 + m] = acc[v];
  }
}

// --- el/er: per (node, head) dot of feat with attention vectors -------------
__global__ void el_er_kernel(const float* __restrict__ feat,
                             const float* __restrict__ attn_l,
                             const float* __restrict__ attn_r,
                             float* __restrict__ el, float* __restrict__ er,
                             int N) {
  int i = blockIdx.x * blockDim.x + threadIdx.x; // over N*NHEAD
  if (i >= N * NHEAD) return;
  const int n = i >> 1, h = i & 1;
  const float* f  = feat + (size_t)n * DOUT + h * HID;
  const float* al = attn_l + h * HID;
  const float* ar = attn_r + h * HID;
  float sl = 0.f, sr = 0.f;
#pragma unroll 8
  for (int c = 0; c < HID; ++c) {
    const float v = f[c];
    sl += v * al[c];
    sr += v * ar[c];
  }
  el[i] = sl;
  er[i] = sr;
}

__device__ __forceinline__ float leaky(float x) {
  return x > 0.f ? x : SLOPE * x;
}

// --- edge pass 1: segment max over dst --------------------------------------
__global__ void edge_max_kernel(const int* __restrict__ src,
                                const int* __restrict__ dst,
                                const float* __restrict__ el,
                                const float* __restrict__ er,
                                float* __restrict__ nmax, int E) {
  int i = blockIdx.x * blockDim.x + threadIdx.x; // over E*NHEAD
  if (i >= E * NHEAD) return;
  const int e = i >> 1, h = i & 1;
  const int s = src[e], d = dst[e];
  const float v = leaky(el[s * NHEAD + h] + er[d * NHEAD + h]);
  atomic_max_f32(&nmax[d * NHEAD + h], v);
}

// --- edge pass 2: ex = exp(e - max[dst]); den[dst] += ex --------------------
__global__ void edge_exp_kernel(const int* __restrict__ src,
                                const int* __restrict__ dst,
                                const float* __restrict__ el,
                                const float* __restrict__ er,
                                const float* __restrict__ nmax,
                                float* __restrict__ exbuf,
                                float* __restrict__ nden, int E) {
  int i = blockIdx.x * blockDim.x + threadIdx.x; // over E*NHEAD
  if (i >= E * NHEAD) return;
  const int e = i >> 1, h = i & 1;
  const int s = src[e], d = dst[e];
  const float v = leaky(el[s * NHEAD + h] + er[d * NHEAD + h]);
  const float ex = __expf(v - nmax[d * NHEAD + h]);
  exbuf[i] = ex;
  atomic_add_f32(&nden[d * NHEAD + h], ex);
}

// --- edge pass 3: accum[dst] += alpha * feat[src] (128 lanes / edge) --------
__global__ void edge_scatter_kernel(const int* __restrict__ src,
                                    const int* __restrict__ dst,
                                    const float* __restrict__ feat,
                                    const float* __restrict__ exbuf,
                                    const float* __restrict__ nden,
                                    float* __restrict__ accum, int E) {
  const int t = threadIdx.x;               // 0..255  -> 2 edges per block
  const int e = blockIdx.x * 2 + (t >> 7);
  if (e >= E) return;
  const int c = t & 127;
  const int h = c >> 6;
  const int s = src[e], d = dst[e];
  float den = nden[d * NHEAD + h];
  den = (den == 0.f) ? 1.f : den;
  const float alpha = exbuf[e * NHEAD + h] / den;
  atomic_add_f32(&accum[(size_t)d * DOUT + c],
                 alpha * feat[(size_t)s * DOUT + c]);
}

// --- relu + mean over heads: h[n, c] = 0.5*(relu(a[n,c]) + relu(a[n,64+c]))--
__global__ void relu_mean_kernel(const float* __restrict__ accum,
                                 float* __restrict__ h, int N) {
  int i = blockIdx.x * blockDim.x + threadIdx.x; // over N*HID
  if (i >= N * HID) return;
  const int n = i >> 6, c = i & 63;
  const float v0 = accum[(size_t)n * DOUT + c];
  const float v1 = accum[(size_t)n * DOUT + HID + c];
  h[i] = 0.5f * (fmaxf(v0, 0.f) + fmaxf(v1, 0.f));
}

// --- gate: gate[n] = h2[n,:] . w_gate + b_gate ------------------------------
__global__ void gate_kernel(const float* __restrict__ h2,
                            const float* __restrict__ w_gate,
                            const float* __restrict__ b_gate,
                            float* __restrict__ gate, int N) {
  int n = blockIdx.x * blockDim.x + threadIdx.x;
  if (n >= N) return;
  const float* hr = h2 + (size_t)n * HID;
  float s = b_gate[0];
#pragma unroll 8
  for (int c = 0; c < HID; ++c) s += hr[c] * w_gate[c];
  gate[n] = s;
}

// --- per-graph softmax: max, exp+sum, normalize -----------------------------
__global__ void graph_max_kernel(const float* __restrict__ gate,
                                 const int* __restrict__ gid,
                                 float* __restrict__ gmax, int N) {
  int n = blockIdx.x * blockDim.x + threadIdx.x;
  if (n >= N) return;
  atomic_max_f32(&gmax[gid[n]], gate[n]);
}

__global__ void graph_exp_kernel(const float* __restrict__ gate,
                                 const int* __restrict__ gid,
                                 const float* __restrict__ gmax,
                                 float* __restrict__ a_out,
                                 float* __restrict__ gden, int N) {
  int n = blockIdx.x * blockDim.x + threadIdx.x;
  if (n >= N) return;
  const int g = gid[n];
  const float ex = __expf(gate[n] - gmax[g]);
  a_out[n] = ex;
  atomic_add_f32(&gden[g], ex);
}

__global__ void graph_norm_kernel(const int* __restrict__ gid,
                                  const float* __restrict__ gden,
                                  float* __restrict__ a_out, int N) {
  int n = blockIdx.x * blockDim.x + threadIdx.x;
  if (n >= N) return;
  float den = gden[gid[n]];
  den = (den == 0.f) ? 1.f : den;
  a_out[n] = a_out[n] / den;
}

// --- hg[g,:] += a[n] * h2[n,:]; graph_ids sorted -> running per-block sums --
__global__ void hg_accum_kernel(const float* __restrict__ h2,
                                const float* __restrict__ a,
                                const int* __restrict__ gid,
                                float* __restrict__ hg, int N, int chunk) {
  const int c = threadIdx.x; // 0..63, owns column c
  const int n0 = blockIdx.x * chunk;
  int n1 = n0 + chunk;
  if (n1 > N) n1 = N;
  float acc = 0.f;
  int gcur = -1;
  for (int n = n0; n < n1; ++n) {
    const int g = gid[n];
    if (g != gcur) {
      if (gcur >= 0) atomic_add_f32(&hg[gcur * HID + c], acc);
      gcur = g;
      acc = 0.f;
    }
    acc += a[n] * h2[(size_t)n * HID + c];
  }
  if (gcur >= 0) atomic_add_f32(&hg[gcur * HID + c], acc);
}

// --- classifier head: a3 = (hg@Wc1+b1)@Wc2+b2; out = sigmoid(a3) ------------
__global__ void classifier_kernel(const float* __restrict__ hg,
                                  const float* __restrict__ Wc1,
                                  const float* __restrict__ bc1,
                                  const float* __restrict__ Wc2,
                                  const float* __restrict__ bc2,
                                  float* __restrict__ out) {
  __shared__ float a2s[NGRAPH * HID];
  const int t = threadIdx.x; // 0..511
  const int g = t >> 6, j = t & 63;
  float s = bc1[j];
#pragma unroll 8
  for (int k = 0; k < HID; ++k) s += hg[g * HID + k] * Wc1[k * HID + j];
  a2s[g * HID + j] = s;
  __syncthreads();
  if (t < NGRAPH * 2) {
    const int g2 = t >> 1, j2 = t & 1;
    float s2 = bc2[j2];
#pragma unroll 8
    for (int k = 0; k < HID; ++k) s2 += a2s[g2 * HID + k] * Wc2[k * 2 + j2];
    out[g2 * 2 + j2] = 1.f / (1.f + __expf(-s2));
  }
}

// ---------------------------------------------------------------------------

static inline int cdiv(int a, int b) { return (a + b - 1) / b; }

extern "C" void kernel_launch(void* const* d_in, const int* in_sizes, int n_in,
                              void* d_out, int out_size, void* d_ws,
                              size_t ws_size, hipStream_t stream) {
  // inputs (setup_inputs order)
  const float* h_n     = (const float*)d_in[0];
  const int*   src     = (const int*)d_in[1];
  const int*   dst     = (const int*)d_in[2];
  const int*   gid     = (const int*)d_in[3];
  const float* W_fc1   = (const float*)d_in[4];
  const float* attn_l1 = (const float*)d_in[5];
  const float* attn_r1 = (const float*)d_in[6];
  const float* W_fc2   = (const float*)d_in[7];
  const float* attn_l2 = (const float*)d_in[8];
  const float* attn_r2 = (const float*)d_in[9];
  const float* w_gate  = (const float*)d_in[10];
  const float* b_gate  = (const float*)d_in[11];
  const float* W_c1    = (const float*)d_in[12];
  const float* b_c1    = (const float*)d_in[13];
  const float* W_c2    = (const float*)d_in[14];
  const float* b_c2    = (const float*)d_in[15];

  const int N = in_sizes[3];   // graph_ids has N elements
  const int E = in_sizes[1];   // src has E elements
  const int IN = in_sizes[0] / N; // 128

  // workspace layout (floats)
  float* ws    = (float*)d_ws;
  float* feat  = ws;                       // N * 128
  float* accum = feat + (size_t)N * DOUT;  // N * 128
  float* hbuf  = accum + (size_t)N * DOUT; // N * 64
  float* h2buf = hbuf + (size_t)N * HID;   // N * 64
  float* el    = h2buf + (size_t)N * HID;  // N * 2
  float* er    = el + (size_t)N * NHEAD;   // N * 2
  float* nmax  = er + (size_t)N * NHEAD;   // N * 2
  float* nden  = nmax + (size_t)N * NHEAD; // N * 2
  float* exbuf = nden + (size_t)N * NHEAD; // E * 2
  float* gate  = exbuf + (size_t)E * NHEAD;// N
  float* gmax  = gate + (size_t)N;         // 8
  float* gden  = gmax + NGRAPH;            // 8

  // output layout: [sigmoid(a3): G*2] [a: N] [hg: G*64]
  float* out_sig = (float*)d_out;
  float* out_a   = out_sig + NGRAPH * 2;
  float* out_hg  = out_a + N;

  const float NEG_INF = -__builtin_inff();
  const dim3 gemm_block(32, 8);

  // ============================ layer 1 ====================================
  gemm_fc_wmma<<<cdiv(N, 16), gemm_block, 0, stream>>>(h_n, W_fc1, feat, N, IN);
  el_er_kernel<<<cdiv(N * NHEAD, 256), 256, 0, stream>>>(feat, attn_l1, attn_r1,
                                                         el, er, N);
  hipMemsetAsync(accum, 0, (size_t)N * DOUT * sizeof(float), stream);
  hipMemsetAsync(nden, 0, (size_t)N * NHEAD * sizeof(float), stream);
  fill_kernel<<<cdiv(N * NHEAD, 256), 256, 0, stream>>>(nmax, NEG_INF, N * NHEAD);

  edge_max_kernel<<<cdiv(E * NHEAD, 256), 256, 0, stream>>>(src, dst, el, er,
                                                            nmax, E);
  edge_exp_kernel<<<cdiv(E * NHEAD, 256), 256, 0, stream>>>(src, dst, el, er,
                                                            nmax, exbuf, nden, E);
  edge_scatter_kernel<<<cdiv(E, 2), 256, 0, stream>>>(src, dst, feat, exbuf,
                                                      nden, accum, E);
  relu_mean_kernel<<<cdiv(N * HID, 256), 256, 0, stream>>>(accum, hbuf, N);

  // ============================ layer 2 ====================================
  gemm_fc_wmma<<<cdiv(N, 16), gemm_block, 0, stream>>>(hbuf, W_fc2, feat, N, HID);
  el_er_kernel<<<cdiv(N * NHEAD, 256), 256, 0, stream>>>(feat, attn_l2, attn_r2,
                                                         el, er, N);
  hipMemsetAsync(accum, 0, (size_t)N * DOUT * sizeof(float), stream);
  hipMemsetAsync(nden, 0, (size_t)N * NHEAD * sizeof(float), stream);
  fill_kernel<<<cdiv(N * NHEAD, 256), 256, 0, stream>>>(nmax, NEG_INF, N * NHEAD);

  edge_max_kernel<<<cdiv(E * NHEAD, 256), 256, 0, stream>>>(src, dst, el, er,
                                                            nmax, E);
  edge_exp_kernel<<<cdiv(E * NHEAD, 256), 256, 0, stream>>>(src, dst, el, er,
                                                            nmax, exbuf, nden, E);
  edge_scatter_kernel<<<cdiv(E, 2), 256, 0, stream>>>(src, dst, feat, exbuf,
                                                      nden, accum, E);
  relu_mean_kernel<<<cdiv(N * HID, 256), 256, 0, stream>>>(accum, h2buf, N);

  // ===================== global attention pooling ==========================
  gate_kernel<<<cdiv(N, 256), 256, 0, stream>>>(h2buf, w_gate, b_gate, gate, N);
  fill_kernel<<<1, 32, 0, stream>>>(gmax, NEG_INF, NGRAPH);
  hipMemsetAsync(gden, 0, NGRAPH * sizeof(float), stream);
  graph_max_kernel<<<cdiv(N, 256), 256, 0, stream>>>(gate, gid, gmax, N);
  graph_exp_kernel<<<cdiv(N, 256), 256, 0, stream>>>(gate, gid, gmax, out_a,
                                                     gden, N);
  graph_norm_kernel<<<cdiv(N, 256), 256, 0, stream>>>(gid, gden, out_a, N);

  hipMemsetAsync(out_hg, 0, NGRAPH * HID * sizeof(float), stream);
  const int chunk = 512;
  hg_accum_kernel<<<cdiv(N, chunk), HID, 0, stream>>>(h2buf, out_a, gid, out_hg,
                                                      N, chunk);

  // ========================= classifier head ===============================
  classifier_kernel<<<1, NGRAPH * HID, 0, stream>>>(out_hg, W_c1, b_c1, W_c2,
                                                    b_c2, out_sig);
}